// GenerateNodes_25262997635872
// MI455X (gfx1250) — compile-verified
//
#include <hip/hip_runtime.h>

// ---------------------------------------------------------------------------
// GenerateNodes as one GEMM: M=16384 (n*t), K=3215 (3200 x-cols + 15 seed
// cols), N=60 (20 cloned Linears x 3 outputs, col j = o*20+k).
//  - fp32 WMMA (V_WMMA_F32_16X16X4_F32): matches fp32 reference precision.
//  - B (W^T) staged into LDS in K-chunks with GLOBAL_LOAD_ASYNC_TO_LDS_B32,
//    double-buffered so staging of chunk i+1 overlaps WMMA compute of chunk i.
// ---------------------------------------------------------------------------

typedef __attribute__((ext_vector_type(2))) float v2f;
typedef __attribute__((ext_vector_type(8))) float v8f;

#define N_ROWS    16384      // N*T
#define C_X       3200       // TOTAL_NODES*C_IN
#define LIN_IN_K  3215       // C_X + NUM_SEEDS*C_OUT
#define SEED_K    15
#define NCOL      60
#define NPAD      64
#define NNODES    20
#define KC        200        // K-chunk staged in LDS (multiple of 4)
#define NCHUNK    (C_X / KC) // 16
#define WAVES     4
#define NTHREADS  (WAVES * 32)

__global__ __launch_bounds__(NTHREADS)
void gen_nodes_gemm(const float* __restrict__ x,
                    const float* __restrict__ seeds,
                    const float* __restrict__ W,
                    const float* __restrict__ bias,
                    float* __restrict__ out)
{
    // B staged as K-pair rows: buf[p*NPAD + j] = (B[2p][j], B[2p+1][j])
    // -> one ds_load_2addr_b64 yields both K values a lane-half needs.
    __shared__ float2 Blds[2][(KC / 2) * NPAD];   // 2 x 50 KB double buffer

    const int tid  = threadIdx.x;
    const int lane = tid & 31;
    const int wave = tid >> 5;
    const int half = lane >> 4;          // K-half select inside a 16x16x4 step
    const int l15  = lane & 15;

    const int tile    = blockIdx.x * WAVES + wave;   // 0..1023
    const int rowBase = tile * 16;

    // ---- staging mapping: thread -> (column j, c parity crow0) ----
    const int  j     = tid & 63;
    const int  crow0 = tid >> 6;                     // 0..1
    const bool jval  = (j < NCOL);
    const int  jj    = jval ? j : 0;   // pad cols get col-0 copies (discarded)
    const size_t wbase = (size_t)(jj % NNODES) * (3u * LIN_IN_K)
                       + (size_t)(jj / NNODES) * LIN_IN_K;

    // Raw LDS byte offsets (low 32 bits of the generic address are the LDS
    // offset per the flat-address aperture rules).
    const unsigned ldsBase0 = (unsigned)(size_t)(void*)&Blds[0][0];
    const unsigned ldsBase1 = (unsigned)(size_t)(void*)&Blds[1][0];
    const unsigned stageOff = (unsigned)(j * 8 + crow0 * 4);

    // Issue this thread's share of one K-chunk of B as async LDS loads.
    // Thread covers c = crow0, crow0+2, ... (100 elements); LDS stride 512 B.
    auto stage_async = [&](int c0, unsigned bufBase) {
        unsigned     loff = bufBase + stageOff;
        const float* gp   = W + wbase + (size_t)c0 + (size_t)crow0;
        #pragma unroll 4
        for (int t = 0; t < KC / 2; ++t) {
            asm volatile("global_load_async_to_lds_b32 %0, %1, off"
                         :: "v"(loff), "v"(gp) : "memory");
            loff += NPAD * 8;   // next K-pair row
            gp   += 2;          // next c of same parity
        }
    };

    // ---- per-lane A-fragment source pointers (ISA 16x4 f32 A layout:
    //      lanes 0-15: K=k,k+1 ; lanes 16-31: K=k+2,k+3; M = lane&15) ----
    const float* xrow = x     + (size_t)(rowBase + l15) * C_X + 2 * half;
    const float* srow = seeds + (size_t)(rowBase + l15) * SEED_K;

    v8f acc[4] = {v8f{}, v8f{}, v8f{}, v8f{}};

    // ---------------- main K loop over the x part (double-buffered) --------
    stage_async(0, ldsBase0);                       // prologue: chunk 0

    for (int ci = 0; ci < NCHUNK; ++ci) {
        asm volatile("s_wait_asynccnt 0" ::: "memory");  // own async writes done
        __syncthreads();                                 // all waves' writes visible

        if (ci + 1 < NCHUNK)                             // prefetch next chunk
            stage_async((ci + 1) * KC, (ci & 1) ? ldsBase0 : ldsBase1);

        const float2* buf = &Blds[ci & 1][0];
        const float*  ap  = xrow + ci * KC;
        #pragma unroll 4
        for (int kk = 0; kk < KC; kk += 4) {
            const float2 av = *(const float2*)(ap + kk);   // global_load_b64
            v2f a; a.x = av.x; a.y = av.y;
            const float2* bp = &buf[((kk >> 1) + half) * NPAD + l15];
            #pragma unroll
            for (int nt = 0; nt < 4; ++nt) {
                float2 bb = bp[nt * 16];                   // ds_load (paired)
                v2f b; b.x = bb.x; b.y = bb.y;
                acc[nt] = __builtin_amdgcn_wmma_f32_16x16x4_f32(
                    false, a, false, b, (short)0, acc[nt], false, false);
            }
        }
    }

    // ---------------- seeds tail: K = 3200..3214 (+1 zero pad) -------------
    {
        __syncthreads();   // last compute (buf 1) done in all waves; reuse buf 0
        for (int c = crow0; c < 16; c += 2) {
            const int cg = C_X + c;
            float v = (cg < LIN_IN_K) ? W[wbase + (size_t)cg] : 0.0f;
            ((float*)&Blds[0][0])[((c >> 1) * NPAD + j) * 2 + (c & 1)] = v;
        }
        __syncthreads();

        #pragma unroll
        for (int kk = 0; kk < 16; kk += 4) {
            const int kl = kk + 2 * half;
            v2f a;
            a.x = (kl + 0 < SEED_K) ? srow[kl + 0] : 0.0f;  // selects, no EXEC div.
            a.y = (kl + 1 < SEED_K) ? srow[kl + 1] : 0.0f;
            const float2* bp = &Blds[0][((kk >> 1) + half) * NPAD + l15];
            #pragma unroll
            for (int nt = 0; nt < 4; ++nt) {
                float2 bb = bp[nt * 16];
                v2f b; b.x = bb.x; b.y = bb.y;
                acc[nt] = __builtin_amdgcn_wmma_f32_16x16x4_f32(
                    false, a, false, b, (short)0, acc[nt], false, false);
            }
        }
    }

    // ---------------- epilogue: bias add + store ---------------------------
    // D layout: VGPR v -> M = v + 8*half, N = l15 (per ISA C/D table)
    #pragma unroll
    for (int nt = 0; nt < 4; ++nt) {
        const int col = nt * 16 + l15;
        if (col < NCOL) {
            const float bv = bias[(col % NNODES) * 3 + (col / NNODES)];
            #pragma unroll
            for (int v = 0; v < 8; ++v) {
                const int m = v + 8 * half;
                out[(size_t)(rowBase + m) * NCOL + col] = acc[nt][v] + bv;
            }
        }
    }
}

extern "C" void kernel_launch(void* const* d_in, const int* in_sizes, int n_in,
                              void* d_out, int out_size, void* d_ws, size_t ws_size,
                              hipStream_t stream) {
    (void)in_sizes; (void)n_in; (void)out_size; (void)d_ws; (void)ws_size;
    const float* x     = (const float*)d_in[0];
    const float* seeds = (const float*)d_in[1];
    const float* W     = (const float*)d_in[2];
    const float* b     = (const float*)d_in[3];
    float* out = (float*)d_out;

    dim3 grid(N_ROWS / (16 * WAVES));   // 1024 M-tiles / 4 waves = 256 blocks
    gen_nodes_gemm<<<grid, NTHREADS, 0, stream>>>(x, seeds, W, b, out);
}